// VQSegmentationModel_67499706024064
// MI455X (gfx1250) — compile-verified
//
#include <hip/hip_runtime.h>
#include <hip/hip_bf16.h>

// ---------------- CDNA5 WMMA fp32 16x16x4 ----------------
typedef float v2f __attribute__((ext_vector_type(2)));
typedef float v8f __attribute__((ext_vector_type(8)));
typedef unsigned int u32x4 __attribute__((ext_vector_type(4)));
typedef int i32x4 __attribute__((ext_vector_type(4)));
typedef int i32x8 __attribute__((ext_vector_type(8)));

__device__ __forceinline__ v8f wmma4(v2f a, v2f b, v8f c) {
  // D(16x16,f32) = A(16x4,f32) * B(4x16,f32) + C
  return __builtin_amdgcn_wmma_f32_16x16x4_f32(false, a, false, b, (short)0, c,
                                               false, false);
}

#if __has_builtin(__builtin_amdgcn_tensor_load_to_lds) && \
    __has_builtin(__builtin_amdgcn_s_wait_tensorcnt)
#define USE_TDM 1
#endif

// ---------------- problem dims ----------------
#define BB    16
#define TT    8000
#define TN    800      // T / CHUNK_SIZE
#define HID   512
#define LAT   128
#define KCB   1024
#define QCB   4
#define WD    5
#define CNT   1599     // (T - 2*Wd)/Wd + 1
#define NS    (BB*CNT) // 25584 decoder samples
#define EPSF  1e-12f

// workspace layout (floats)
#define OFF_Z1    ((size_t)0)               // 16*512*800     = 6,553,600
#define OFF_ZUP   ((size_t)6553600)         // 16*512*8000    = 65,536,000 (reused as h)
#define OFF_Z2    (OFF_ZUP + 65536000ull)   // 16*512*8000    = 65,536,000 (reused as dec_in)
#define OFF_ZE    (OFF_Z2  + 65536000ull)   // 128000*128     = 16,384,000
#define OFF_QOUT  (OFF_ZE  + 16384000ull)   // 128000*128     = 16,384,000
#define OFF_CC    (OFF_QOUT+ 16384000ull)   // 4*1024
#define OFF_ACC   (OFF_CC  + 4096ull)       // 16 accumulators

// ==========================================================
// TDM staging of a 16 x CIN weight slab for conv tap dk:
//   ldsW[m*cin + ci] = wsrc[((co0+m)*cin + ci)*kw + dk]
// D# 3D tile: x-run = 1 element (runs are strided by kw),
//             y = cin rows (tensor_dim0_stride = kw),
//             z = 16 channels (tensor_dim1_stride = cin*kw).
// LDS written contiguously in x,y,z order => matches [m][ci] layout.
// ==========================================================
__device__ __forceinline__ void stage_slab(float* ldsW, const float* wsrc,
                                           int cin, int kw, int co0, int dk) {
  __syncthreads();
#ifdef USE_TDM
  if (threadIdx.x < 32) {  // wave 0 issues the DMA (TDM ignores EXEC)
    unsigned long long ga =
        (unsigned long long)(size_t)(wsrc + ((size_t)co0 * cin) * kw + dk);
    unsigned lds = (unsigned)(size_t)ldsW;  // low 32 bits = LDS byte offset
    unsigned s1 = (unsigned)(cin * kw);
    u32x4 g0;
    g0[0] = 1u;                                    // count=1 (valid user D#)
    g0[1] = lds;                                   // lds_addr (bytes)
    g0[2] = (unsigned)(ga & 0xffffffffull);        // global_addr[31:0]
    g0[3] = (unsigned)((ga >> 32) & 0x01ffffffull) // global_addr[56:32]
            | (2u << 30);                          // type = 2 ("image")
    i32x8 g1;
    g1[0] = (int)(2u << 16);                       // data_size = 4 bytes
    g1[1] = (int)(1u << 16);                       // tensor_dim0 = 1
    g1[2] = (int)(((unsigned)cin & 0xffffu) << 16);// tensor_dim1 lo16 = cin
    g1[3] = (int)(1u << 16);                       // tile_dim0 = 1
    g1[4] = (int)((16u << 16) | (unsigned)cin);    // tile_dim2=16 | tile_dim1=cin
    g1[5] = (int)(unsigned)kw;                     // tensor_dim0_stride[31:0]
    g1[6] = (int)((s1 & 0xffffu) << 16);           // dim1_stride[15:0] in hi16
    g1[7] = (int)(s1 >> 16);                       // dim1_stride[47:16]
    i32x4 g2;
    g2[0] = 16;                                    // tensor_dim2
    g2[1] = 1;                                     // tensor_dim3 (tile_dim3=0)
    g2[2] = 0;                                     // tensor_dim2_stride lo
    g2[3] = 0;                                     // stride hi | tile_dim3=0
    i32x4 g3 = {0, 0, 0, 0};
    i32x8 g4 = {0, 0, 0, 0, 0, 0, 0, 0};
    __builtin_amdgcn_tensor_load_to_lds(g0, g1, g2, g3, g4, 0);
    __builtin_amdgcn_s_wait_tensorcnt(0);
  }
#else
  for (int i = threadIdx.x; i < 16 * cin; i += 256)
    ldsW[i] = wsrc[((size_t)(co0 + i / cin) * cin + (i % cin)) * kw + dk];
#endif
  __syncthreads();
}

// ==========================================================
// 0) init: zero accumulators, codebook squared norms
// ==========================================================
__global__ void init_kernel(const float* __restrict__ cb, float* __restrict__ cc,
                            float* __restrict__ acc) {
  int i = blockIdx.x * blockDim.x + threadIdx.x;
  if (i < 16) acc[i] = 0.f;
  if (i < QCB * KCB) {
    const float* c = cb + (size_t)i * LAT;
    float s = 0.f;
    for (int j = 0; j < LAT; ++j) s += c[j] * c[j];
    cc[i] = s;
  }
}

// ==========================================================
// 1) chunk-reshape + enc1 conv(k=3, Cin=20) + ReLU
// ==========================================================
__global__ void enc1_kernel(const float* __restrict__ traj, const float* __restrict__ w1,
                            const float* __restrict__ b1, float* __restrict__ z1) {
  int idx = blockIdx.x * blockDim.x + threadIdx.x;
  if (idx >= BB * HID * TN) return;
  int t = idx % TN;
  int bc = idx / TN;
  int co = bc % HID;
  int b  = bc / HID;
  float s = b1[co];
  #pragma unroll
  for (int dk = 0; dk < 3; ++dk) {
    int tp = t + dk - 1;
    float msk = (tp >= 0 && tp < TN) ? 1.f : 0.f;
    int tc = min(max(tp, 0), TN - 1);
    const float* wp = w1 + (size_t)co * 20 * 3 + dk;
    #pragma unroll
    for (int d = 0; d < 2; ++d) {
      const float* xp = traj + ((size_t)b * 2 + d) * TT + (size_t)tc * 10;
      #pragma unroll
      for (int c = 0; c < 10; ++c)
        s += msk * wp[(d * 10 + c) * 3] * xp[c];
    }
  }
  z1[idx] = fmaxf(s, 0.f);
}

// ==========================================================
// 2) adaptive_avg_pool(800->4000) + linear interp(4000->8000), fused.
// ==========================================================
__global__ void upsample_kernel(const float* __restrict__ z1, float* __restrict__ zup) {
  size_t idx = (size_t)blockIdx.x * blockDim.x + threadIdx.x;
  if (idx >= (size_t)BB * HID * TT) return;
  int j = (int)(idx % TT);
  size_t bc = idx / TT;
  float src = fmaxf((j + 0.5f) * 0.5f - 0.5f, 0.f);
  int i0 = (int)src;
  int i1 = min(i0 + 1, 3999);
  float fr = src - (float)i0;
  const float* p = z1 + bc * TN;
  zup[idx] = p[i0 / 5] * (1.f - fr) + p[i1 / 5] * fr;
}

// ==========================================================
// 3/4) conv over time as GEMM with fp32 WMMA, TDM-staged weights,
//      double-buffered fragment pipeline.
// ==========================================================
template <int CIN, int KW, bool RELU, bool TO_TC>
__global__ __launch_bounds__(256) void conv_time_kernel(
    const float* __restrict__ in, const float* __restrict__ wgt,
    const float* __restrict__ bias, float* __restrict__ out, int cout_tot) {
  __shared__ float ldsW[16 * CIN];
  const int tid = threadIdx.x, lane = tid & 31, wv = tid >> 5;
  const int n16 = lane & 15, kh = lane >> 4;
  const int co0 = blockIdx.x * 16;
  const int b = blockIdx.y;
  const float* zb = in + (size_t)b * CIN * TT;
  const int NTILE = TT / 16;            // 500
  const int ITERS = (NTILE + 7) / 8;    // 63

  for (int it = 0; it < ITERS; ++it) {
    int t_tile = it * 8 + wv;
    v8f acc = {};
    for (int dk = 0; dk < KW; ++dk) {
      stage_slab(ldsW, wgt, CIN, KW, co0, dk);
      if (t_tile < NTILE) {
        int tt = t_tile * 16 + n16 + dk - KW / 2;
        float msk = (tt >= 0 && tt < TT) ? 1.f : 0.f;
        int ts = min(max(tt, 0), TT - 1);
        const float* lw = &ldsW[n16 * CIN];
        const float* zcol = zb + ts;
        v2f a0, b0;
        {
          int k2 = 2 * kh;
          a0.x = lw[k2];  a0.y = lw[k2 + 1];
          b0.x = msk * zcol[(size_t)k2 * TT];
          b0.y = msk * zcol[(size_t)(k2 + 1) * TT];
        }
        #pragma unroll 4
        for (int ci = 0; ci < CIN; ci += 4) {
          v2f a1 = a0, b1 = b0;
          if (ci + 4 < CIN) {
            int k2 = ci + 4 + 2 * kh;
            a1.x = lw[k2];  a1.y = lw[k2 + 1];
            b1.x = msk * zcol[(size_t)k2 * TT];
            b1.y = msk * zcol[(size_t)(k2 + 1) * TT];
          }
          acc = wmma4(a0, b0, acc);
          a0 = a1; b0 = b1;
        }
      }
    }
    if (t_tile < NTILE) {
      int t = t_tile * 16 + n16;
      #pragma unroll
      for (int v = 0; v < 8; ++v) {
        int co = co0 + v + 8 * kh;
        float r = acc[v] + bias[co];
        if (RELU) r = fmaxf(r, 0.f);
        if (TO_TC) out[((size_t)b * TT + t) * cout_tot + co] = r;
        else       out[((size_t)b * cout_tot + co) * TT + t] = r;
      }
    }
  }
}

// ==========================================================
// 5) residual VQ: one wave per 16-row group, 4 stages in-kernel.
// ==========================================================
__global__ __launch_bounds__(256) void vq_kernel(
    const float* __restrict__ ze, const float* __restrict__ cb,
    const float* __restrict__ cc, float* __restrict__ qout,
    float* __restrict__ codes_out, float* __restrict__ acc) {
  extern __shared__ float sh[];
  const int tid = threadIdx.x, lane = tid & 31, wv = tid >> 5;
  const int n16 = lane & 15, kh = lane >> 4;
  float* R  = &sh[(size_t)wv * 4128];  // residual rows 16x128
  float* QA = R + 2048;                // quantized-out accumulator 16x128
  float* RR = QA + 2048;               // |r|^2 per row
  int*   SI = (int*)(RR + 16);         // argmin index per row
  const size_t g = (size_t)blockIdx.x * 8 + wv;  // row group 0..7999

  for (int i = lane; i < 2048; i += 32) {
    R[i]  = ze[g * 2048 + i];
    QA[i] = 0.f;
  }
  __syncthreads();

  for (int qi = 0; qi < QCB; ++qi) {
    { // |r|^2 per row
      float s = 0.f;
      const float* rp = &R[n16 * 128 + kh * 64];
      for (int j = 0; j < 64; ++j) s += rp[j] * rp[j];
      s += __shfl_xor(s, 16, 32);
      if (kh == 0) RR[n16] = s;
    }
    __syncthreads();
    float rr8[8];
    #pragma unroll
    for (int v = 0; v < 8; ++v) rr8[v] = RR[v + 8 * kh];

    float bd[8]; int bn[8];
    #pragma unroll
    for (int v = 0; v < 8; ++v) { bd[v] = 3.4e38f; bn[v] = 0; }

    const float* lr = &R[n16 * 128];
    for (int nt = 0; nt < KCB / 16; ++nt) {
      int n = nt * 16 + n16;
      float ccv = cc[qi * KCB + n];
      const float* cbn = cb + ((size_t)qi * KCB + n) * LAT;
      v8f dacc = {};
      v2f a0, b0;
      {
        int k2 = 2 * kh;
        a0.x = lr[k2];  a0.y = lr[k2 + 1];
        b0.x = cbn[k2]; b0.y = cbn[k2 + 1];
      }
      #pragma unroll
      for (int k = 0; k < LAT; k += 4) {
        v2f a1 = a0, b1 = b0;
        if (k + 4 < LAT) {
          int k2 = k + 4 + 2 * kh;
          a1.x = lr[k2];  a1.y = lr[k2 + 1];
          b1.x = cbn[k2]; b1.y = cbn[k2 + 1];
        }
        dacc = wmma4(a0, b0, dacc);
        a0 = a1; b0 = b1;
      }
      #pragma unroll
      for (int v = 0; v < 8; ++v) {
        float d = rr8[v] - 2.f * dacc[v] + ccv;
        if (d < bd[v] || (d == bd[v] && n < bn[v])) { bd[v] = d; bn[v] = n; }
      }
    }
    // argmin reduce over the 16 lanes of each half-wave (n-dimension)
    #pragma unroll
    for (int v = 0; v < 8; ++v) {
      float d = bd[v]; int n = bn[v];
      #pragma unroll
      for (int off = 1; off < 16; off <<= 1) {
        float od = __shfl_xor(d, off, 32);
        int   on = __shfl_xor(n, off, 32);
        if (od < d || (od == d && on < n)) { d = od; n = on; }
      }
      if (n16 == 0) {
        int m = v + 8 * kh;
        SI[m] = n;
        codes_out[(g * 16 + m) * QCB + qi] = (float)n;
      }
    }
    __syncthreads();

    // rotate_to + residual/qout update, one row at a time (all 32 lanes)
    float lossp = 0.f;
    for (int m = 0; m < 16; ++m) {
      const float* q = cb + ((size_t)qi * KCB + SI[m]) * LAT;
      float* rm = &R[m * 128];
      float* qm = &QA[m * 128];
      float rj[4], qj[4];
      #pragma unroll
      for (int j = 0; j < 4; ++j) { rj[j] = rm[lane * 4 + j]; qj[j] = q[lane * 4 + j]; }
      float rq = 0.f, qq = 0.f, lp = 0.f;
      #pragma unroll
      for (int j = 0; j < 4; ++j) {
        rq += rj[j] * qj[j];
        qq += qj[j] * qj[j];
        float e = qj[j] - rj[j];
        lp += e * e;
      }
      #pragma unroll
      for (int off = 1; off < 32; off <<= 1) {
        rq += __shfl_xor(rq, off, 32);
        qq += __shfl_xor(qq, off, 32);
        lp += __shfl_xor(lp, off, 32);
      }
      float rr = RR[m];
      float ns = sqrtf(rr), ntn = sqrtf(qq);
      float invs = 1.f / (ns + EPSF), invt = 1.f / (ntn + EPSF);
      float uu = rr * invs * invs;
      float uq = rq * invs * invt;
      float qh2 = qq * invt * invt;
      float nw = sqrtf(uu + 2.f * uq + qh2);
      float invw = 1.f / (nw + EPSF);
      float rw = (rr * invs + rq * invt) * invw;  // r . w
      float ru = rr * invs;                       // r . u
      float scale = ntn * invs;                   // nt/(ns+eps)
      #pragma unroll
      for (int j = 0; j < 4; ++j) {
        float wj = (rj[j] * invs + qj[j] * invt) * invw;
        float rot = (rj[j] - 2.f * rw * wj + 2.f * ru * qj[j] * invt) * scale;
        qm[lane * 4 + j] += rot;
        rm[lane * 4 + j] = rj[j] - rot;
      }
      if (lane == 0) lossp += lp;
    }
    if (lane == 0) atomicAdd(&acc[qi], lossp);
    __syncthreads();
  }
  for (int i = lane; i < 2048; i += 32) qout[g * 2048 + i] = QA[i];
}

// ==========================================================
// 6) assemble decoder input (NS, 132, 5) from quant + codes
// ==========================================================
__global__ void decin_kernel(const float* __restrict__ qout,
                             const float* __restrict__ codes,
                             float* __restrict__ decin) {
  int idx = blockIdx.x * blockDim.x + threadIdx.x;
  if (idx >= NS * 132 * WD) return;
  int w = idx % WD;
  int sc = idx / WD;
  int ch = sc % 132;
  int s = sc / 132;
  int b = s / CNT, ci = s % CNT;
  int t = ci * WD + w;
  float v;
  if (ch < LAT) v = qout[((size_t)b * TT + t) * LAT + ch];
  else          v = codes[((size_t)b * TT + t) * QCB + (ch - LAT)];
  decin[idx] = v;
}

// ==========================================================
// 7/8) decoder convs as GEMM with fp32 WMMA (LOSS variant fuses recon loss)
// ==========================================================
template <int CIN, int KW, bool RELU, bool LOSS>
__global__ __launch_bounds__(256) void conv_dec_kernel(
    const float* __restrict__ in, const float* __restrict__ wgt,
    const float* __restrict__ bias, float* __restrict__ out,
    const float* __restrict__ qout, float* __restrict__ acc, int cout_tot) {
  __shared__ float ldsW[16 * CIN];
  const int tid = threadIdx.x, lane = tid & 31, wv = tid >> 5;
  const int n16 = lane & 15, kh = lane >> 4;
  const int co0 = blockIdx.x * 16;
  const int NCT = (NS * WD) / 16;  // 7995 column tiles (exact)
  const int slots = gridDim.y * 8;
  const int slot = blockIdx.y * 8 + wv;
  const int ITERS = (NCT + slots - 1) / slots;
  float lsum = 0.f;

  for (int it = 0; it < ITERS; ++it) {
    int ct = it * slots + slot;
    bool active = ct < NCT;
    int col = ct * 16 + n16;
    int s = col / WD, w = col % WD;
    v8f accv = {};
    for (int dk = 0; dk < KW; ++dk) {
      stage_slab(ldsW, wgt, CIN, KW, co0, dk);
      if (active) {
        int wp = w + dk - KW / 2;
        float msk = (wp >= 0 && wp < WD) ? 1.f : 0.f;
        int ws_ = min(max(wp, 0), WD - 1);
        const float* base = in + (size_t)s * CIN * WD + ws_;
        const float* lw = &ldsW[n16 * CIN];
        v2f a0, b0;
        {
          int k2 = 2 * kh;
          a0.x = lw[k2];  a0.y = lw[k2 + 1];
          b0.x = msk * base[(size_t)k2 * WD];
          b0.y = msk * base[(size_t)(k2 + 1) * WD];
        }
        #pragma unroll 4
        for (int ci = 0; ci < CIN; ci += 4) {
          v2f a1 = a0, b1 = b0;
          if (ci + 4 < CIN) {
            int k2 = ci + 4 + 2 * kh;
            a1.x = lw[k2];  a1.y = lw[k2 + 1];
            b1.x = msk * base[(size_t)k2 * WD];
            b1.y = msk * base[(size_t)(k2 + 1) * WD];
          }
          accv = wmma4(a0, b0, accv);
          a0 = a1; b0 = b1;
        }
      }
    }
    if (active) {
      if (LOSS) {
        int b = s / CNT, ci_ = s % CNT;
        int t = WD + ci_ * WD + w;
        #pragma unroll
        for (int v = 0; v < 8; ++v) {
          int co = co0 + v + 8 * kh;
          float r = accv[v] + bias[co];
          float e = r - qout[((size_t)b * TT + t) * LAT + co];
          lsum += e * e;
        }
      } else {
        #pragma unroll
        for (int v = 0; v < 8; ++v) {
          int co = co0 + v + 8 * kh;
          float r = accv[v] + bias[co];
          if (RELU) r = fmaxf(r, 0.f);
          out[((size_t)s * cout_tot + co) * WD + w] = r;
        }
      }
    }
  }
  if (LOSS) {
    #pragma unroll
    for (int off = 1; off < 32; off <<= 1) lsum += __shfl_xor(lsum, off, 32);
    if (lane == 0) atomicAdd(&acc[4], lsum);
  }
}

// ==========================================================
// 9) smoothness loss on first-codebook codes
// ==========================================================
__global__ void smooth_kernel(const float* __restrict__ codes, float* __restrict__ acc) {
  int idx = blockIdx.x * blockDim.x + threadIdx.x;
  float diff = 0.f;
  if (idx < BB * (TT - 1)) {
    int b = idx / (TT - 1), t = idx % (TT - 1);
    float a = codes[((size_t)b * TT + t) * QCB];
    float c = codes[((size_t)b * TT + t + 1) * QCB];
    diff = (a != c) ? 1.f : 0.f;
  }
  #pragma unroll
  for (int off = 1; off < 32; off <<= 1) diff += __shfl_xor(diff, off, 32);
  if ((threadIdx.x & 31) == 0 && diff != 0.f) atomicAdd(&acc[5], diff);
}

// ==========================================================
// 10) finalize scalar losses
// ==========================================================
__global__ void finalize_kernel(const float* __restrict__ acc, float* __restrict__ out) {
  if (threadIdx.x == 0 && blockIdx.x == 0) {
    float vq = 0.f;
    for (int qi = 0; qi < QCB; ++qi)
      vq += acc[qi] / ((float)BB * TT * LAT);
    vq *= 0.25f;
    float recon = acc[4] / ((float)NS * LAT * WD);
    float smooth = acc[5] / ((float)BB * (TT - 1));
    out[512000] = recon * 1.0f + vq * 1.0f + smooth * 0.1f;
    out[512001] = recon;
    out[512002] = vq;
    out[512003] = smooth;
  }
}

// ==========================================================
extern "C" void kernel_launch(void* const* d_in, const int* in_sizes, int n_in,
                              void* d_out, int out_size, void* d_ws, size_t ws_size,
                              hipStream_t stream) {
  (void)in_sizes; (void)n_in; (void)out_size; (void)ws_size;
  const float* traj = (const float*)d_in[0];
  // d_in[1] = masks (unused by the reference math)
  const float* w1   = (const float*)d_in[2];
  const float* b1   = (const float*)d_in[3];
  const float* w2a  = (const float*)d_in[4];
  const float* b2a  = (const float*)d_in[5];
  const float* w2b  = (const float*)d_in[6];
  const float* b2b  = (const float*)d_in[7];
  const float* dw1  = (const float*)d_in[8];
  const float* db1  = (const float*)d_in[9];
  const float* dw2  = (const float*)d_in[10];
  const float* db2  = (const float*)d_in[11];
  const float* cb   = (const float*)d_in[12];
  float* out = (float*)d_out;

  float* ws   = (float*)d_ws;
  float* z1   = ws + OFF_Z1;
  float* zup  = ws + OFF_ZUP;   // reused later as h
  float* z2   = ws + OFF_Z2;    // reused later as dec_in
  float* ze   = ws + OFF_ZE;
  float* qout = ws + OFF_QOUT;
  float* cc   = ws + OFF_CC;
  float* acc  = ws + OFF_ACC;
  float* h     = zup;
  float* decin = z2;

  init_kernel<<<16, 256, 0, stream>>>(cb, cc, acc);
  enc1_kernel<<<(BB * HID * TN + 255) / 256, 256, 0, stream>>>(traj, w1, b1, z1);
  upsample_kernel<<<(int)(((size_t)BB * HID * TT + 255) / 256), 256, 0, stream>>>(z1, zup);
  conv_time_kernel<HID, 5, true, false>
      <<<dim3(HID / 16, BB), 256, 0, stream>>>(zup, w2a, b2a, z2, HID);
  conv_time_kernel<HID, 7, false, true>
      <<<dim3(LAT / 16, BB), 256, 0, stream>>>(z2, w2b, b2b, ze, LAT);
  vq_kernel<<<1000, 256, 8 * 4128 * sizeof(float), stream>>>(ze, cb, cc, qout, out, acc);
  decin_kernel<<<(NS * 132 * WD + 255) / 256, 256, 0, stream>>>(qout, out, decin);
  conv_dec_kernel<132, 5, true, false>
      <<<dim3(HID / 16, 100), 256, 0, stream>>>(decin, dw1, db1, h, nullptr, acc, HID);
  conv_dec_kernel<HID, 3, false, true>
      <<<dim3(LAT / 16, 100), 256, 0, stream>>>(h, dw2, db2, nullptr, qout, acc, LAT);
  smooth_kernel<<<(BB * (TT - 1) + 255) / 256, 256, 0, stream>>>(out, acc);
  finalize_kernel<<<1, 64, 0, stream>>>(acc, out);
}